// FormantSysth_33621003993673
// MI455X (gfx1250) — compile-verified
//
#include <hip/hip_runtime.h>
#include <hip/hip_bf16.h>
#include <stdint.h>

#ifndef __has_builtin
#define __has_builtin(x) 0
#endif

#define T_LEN   2048
#define NM      64
#define KH      30
#define NF      6
#define TPB     4                 // t-columns per 256-thread block
#define RAW_N   152               // 4 f0 + 4 loud + 8 amps + 24 fr + 24 bwf + 24 af + 64 timbre

typedef __attribute__((address_space(1))) int  gint_t;   // global
typedef __attribute__((address_space(3))) int  lint_t;   // LDS

__device__ __forceinline__ float rcp_f (float x) { return __builtin_amdgcn_rcpf(x);  }  // v_rcp_f32
__device__ __forceinline__ float exp2_f(float x) { return __builtin_amdgcn_exp2f(x); }  // v_exp_f32
__device__ __forceinline__ float log2_f(float x) { return __builtin_amdgcn_logf(x);  }  // v_log_f32

__global__ __launch_bounds__(256)
void formant_synth_kernel(const float* __restrict__ f0,
                          const float* __restrict__ loud,
                          const float* __restrict__ amps,
                          const float* __restrict__ frf,
                          const float* __restrict__ bwf,
                          const float* __restrict__ aff,
                          const float* __restrict__ timbre,
                          float* __restrict__ out)
{
    __shared__ float                 raw[RAW_N];       // staged block inputs (async DMA target)
    __shared__ __align__(16) float2  sh_hk[TPB * KH];  // {f0m*inv15, inv15} per (t,k)
    __shared__ float4                sh_fm[TPB * NF];  // {c2, c1, c0, af}: exp2 arg quadratic in m
    __shared__ int                   sh_cnt[TPB * 2];  // active-harmonic count per (t, half-wave)

    const int tid = (int)threadIdx.x;
    const int b   = (int)(blockIdx.x >> 9);            // T_LEN / TPB = 512 blocks per batch
    const int t0  = (int)(blockIdx.x & 511) * TPB;

    if (tid < TPB * 2) sh_cnt[tid] = 0;

    // ------- stage per-block inputs into LDS via async global->LDS DMA -------
    if (tid < RAW_N) {
        const int j   = tid;
        const int ja  = j - 8;                                   // amps segment
        const int jf  = (j - 16) - ((j - 16) / 24) * 24;         // formant local idx
        const int ff  = jf % NF;                                 // slot = 16+seg*24+tt*6+ff
        const int ftt = jf / NF;

        const float* base =
            (j <  4) ? f0   :
            (j <  8) ? loud :
            (j < 16) ? amps :
            (j < 40) ? frf  :
            (j < 64) ? bwf  :
            (j < 88) ? aff  : timbre;

        const int off =
            (j <  4) ? (b * T_LEN + t0 + j)                          :
            (j <  8) ? (b * T_LEN + t0 + (j - 4))                    :
            (j < 16) ? ((b * 2 + (ja >> 2)) * T_LEN + t0 + (ja & 3)) :
            (j < 88) ? ((b * NF + ff) * T_LEN + t0 + ftt)            :
                       (j - 88);

        const float* gp = base + off;
#if __has_builtin(__builtin_amdgcn_global_load_async_to_lds_b32)
        __builtin_amdgcn_global_load_async_to_lds_b32(
            (gint_t*)(uintptr_t)gp,
            (lint_t*)(uint32_t)(uintptr_t)&raw[tid],
            0, 0);
#else
        raw[tid] = *gp;
#endif
    }
#if __has_builtin(__builtin_amdgcn_s_wait_asynccnt)
    __builtin_amdgcn_s_wait_asynccnt(0);
#else
    asm volatile("s_wait_asynccnt 0" ::: "memory");
#endif
    __syncthreads();

    // ---------------- phase 1: per-(t,k) harmonic params ----------------------
    if (tid < TPB * KH) {
        const int   tt    = tid / KH;
        const int   kk    = tid - tt * KH;
        const float MEL_K = 12.19047619f;                       // 24*64/126
        const float f0v   = raw[tt];
        const float f0hz  = 440.0f * exp2_f(__builtin_fmaf(f0v, 5.25f, -1.29166667f));
        const float f0k   = f0hz * (float)(kk + 1);
        const float bwhz  = __builtin_fmaf(f0k, 0.10793900f, 24.7f);  // 24.7*(1+0.00437*f0k)
        const float lo    = fmaxf(__builtin_fmaf(bwhz, -0.5f, f0k), 0.001f);
        const float hi    = __builtin_fmaf(bwhz,  0.5f, f0k);
        const float bwm   = MEL_K * (log2_f(hi) - log2_f(lo));
        const float f0m   = (log2_f(f0k) - 7.48969305f) * MEL_K;      // mel(f0k)
        const float inv15 = rcp_f(1.5f * bwm);
        sh_hk[tid] = make_float2(f0m * inv15, inv15);           // r = fma(m, inv, -f0m*inv)

        // window lower edge; harmonic is exactly zero for all m <= M if edge > M
        const float edge = __builtin_fmaf(-1.5f, bwm, f0m);
        if (edge <= 63.0f) atomicMax(&sh_cnt[tt * 2 + 1], kk + 1);
        if (edge <= 31.0f) atomicMax(&sh_cnt[tt * 2 + 0], kk + 1);
    }
    // ---------------- phase 1b: per-(t,F) formant params ----------------------
    if (tid < TPB * NF) {
        const float frv = raw[16 + tid] * 64.0f;
        const float bv  = raw[40 + tid] * 64.0f;
        const float n   = -0.72134752f * rcp_f(bv * bv);        // -log2(e)/(2*bwf^2)
        // exp2 argument = n*(m-fr)^2 = c2*m^2 + c1*m + c0
        sh_fm[tid] = make_float4(n, -2.0f * n * frv, n * frv * frv, raw[64 + tid]);
    }
    __syncthreads();

    // ---------------- phase 2: one output element per thread ------------------
    const int   tt = tid >> 6;
    const int   m  = tid & 63;
    const float fm = (float)m;

    // trip count is uniform per wave (each wave = one (column, half) pair)
    const int kc = __builtin_amdgcn_readfirstlane(sh_cnt[tt * 2 + (m >> 5)]);

    // two harmonics per ds_load_b128 (tt*KH*8 bytes is 16B-aligned), dual accumulators
    float hsum0 = 0.0f, hsum1 = 0.0f;
    const float4* __restrict__ ph4 = (const float4*)&sh_hk[tt * KH];
    const int k2 = kc >> 1;
    for (int i = 0; i < k2; ++i) {
        const float4 h  = ph4[i];                               // {b0,a0,b1,a1}
        const float  r0 = __builtin_fmaf(fm, h.y, -h.x);
        const float  r1 = __builtin_fmaf(fm, h.w, -h.z);
        hsum0 += fmaxf(__builtin_fmaf(-r0, r0, 1.0f), 0.0f);
        hsum1 += fmaxf(__builtin_fmaf(-r1, r1, 1.0f), 0.0f);
    }
    if (kc & 1) {                                               // wave-uniform branch
        const float2 h = sh_hk[tt * KH + (kc - 1)];
        const float  r = __builtin_fmaf(fm, h.y, -h.x);
        hsum0 += fmaxf(__builtin_fmaf(-r, r, 1.0f), 0.0f);
    }
    float hsum = (hsum0 + hsum1) * raw[88 + m];                 // * timbre[m]

    const float ld  = raw[4  + tt];
    const float a0  = raw[8  + tt];
    const float a1  = raw[12 + tt];
    const float exc = ld * __builtin_fmaf(a0, hsum, a1);        // loud*(a0*harm + a1*noise)

    float msum = 0.0f;
    const float4* __restrict__ pm = &sh_fm[tt * NF];
#pragma unroll
    for (int f = 0; f < NF; ++f) {
        const float4 q = pm[f];                                 // ds_load_b128, broadcast
        const float  arg = __builtin_fmaf(fm, __builtin_fmaf(fm, q.x, q.y), q.z);
        msum = __builtin_fmaf(q.w, exp2_f(arg), msum);
    }

    out[((size_t)(b * T_LEN + t0 + tt) << 6) + m] = __builtin_fmaf(exc, msum, -1.0f);
}

extern "C" void kernel_launch(void* const* d_in, const int* in_sizes, int n_in,
                              void* d_out, int out_size, void* d_ws, size_t ws_size,
                              hipStream_t stream) {
    const float* f0   = (const float*)d_in[0];   // [B,1,T]
    const float* loud = (const float*)d_in[1];   // [B,1,T]
    const float* amps = (const float*)d_in[2];   // [B,2,T]
    const float* frf  = (const float*)d_in[3];   // [B,F,T]
    const float* bwf  = (const float*)d_in[4];   // [B,F,T]
    const float* aff  = (const float*)d_in[5];   // [B,F,T]
    const float* tim  = (const float*)d_in[6];   // [1,1,64]
    float*       out  = (float*)d_out;           // [B,1,T,64]

    const int BT     = in_sizes[0];              // B*T (= 32768)
    const int blocks = BT / TPB;                 // 8192 blocks of 256 threads
    formant_synth_kernel<<<blocks, 256, 0, stream>>>(f0, loud, amps, frf, bwf, aff, tim, out);
}